// EquivariantProductBasisBlock_25082609009192
// MI455X (gfx1250) — compile-verified
//
#include <hip/hip_runtime.h>

typedef float v2f __attribute__((ext_vector_type(2)));
typedef float v8f __attribute__((ext_vector_type(8)));

#define NN 2048
#define CC 128
#define SS 10
#define DD 9
#define KROWS 96      // padded rows of M (81 real (o,i) rows)
#define KCOLS 56      // padded cols (45 sym-quad + 9 lin + 1 const + 1 pad)
#define MAXT 160      // max node tiles (<= 128 + 10)

// symmetric pair enumeration: col -> (j,l), j<=l
__constant__ int PJ[45] = {0,0,0,0,0,0,0,0,0, 1,1,1,1,1,1,1,1, 2,2,2,2,2,2,2,
                           3,3,3,3,3,3, 4,4,4,4,4, 5,5,5,5, 6,6,6, 7,7, 8};
__constant__ int PL[45] = {0,1,2,3,4,5,6,7,8, 1,2,3,4,5,6,7,8, 2,3,4,5,6,7,8,
                           3,4,5,6,7,8, 4,5,6,7,8, 5,6,7,8, 6,7,8, 7,8, 8};

static __device__ inline v8f wmma4(v2f a, v2f b, v8f c) {
  // V_WMMA_F32_16X16X4_F32 : D(16x16,f32) = A(16x4,f32) * B(4x16,f32) + C
  return __builtin_amdgcn_wmma_f32_16x16x4_f32(false, a, false, b,
                                               (short)0, c, false, false);
}

// ---------------- setup: zero control region ----------------
__global__ void init_kernel(int* ibase) {
  // ints [4096, 4624): counts(16) offsets(16) cursors(16) tspec(160) tstart(160) tcnt(160)
  for (int idx = threadIdx.x; idx < 528; idx += 256) ibase[4096 + idx] = 0;
}

// ---------------- species = argmax(one-hot), histogram ----------------
__global__ void species_kernel(const float* __restrict__ attrs,
                               int* __restrict__ spec, int* __restrict__ counts) {
  int b = blockIdx.x * 256 + threadIdx.x;
  if (b >= NN) return;
  const float* a = attrs + b * SS;
  int s = 0;
  for (int k = 1; k < SS; k++) if (a[k] > a[s]) s = k;
  spec[b] = s;
  atomicAdd(&counts[s], 1);
}

// ---------------- scan + tile descriptor build (single thread) ----------------
__global__ void scan_kernel(int* ibase) {
  if (threadIdx.x != 0) return;
  int* counts  = ibase + 4096;
  int* offsets = ibase + 4112;
  int* cursors = ibase + 4128;
  int* tspec   = ibase + 4144;
  int* tstart  = ibase + 4304;
  int* tcnt    = ibase + 4464;
  int off = 0, nt = 0;
  for (int e = 0; e < SS; e++) {
    offsets[e] = off; cursors[e] = off;
    int ne = counts[e];
    for (int t0 = 0; t0 < ne; t0 += 16) {
      tspec[nt] = e; tstart[nt] = off + t0;
      int r = ne - t0; tcnt[nt] = (r < 16) ? r : 16;
      nt++;
    }
    off += ne;
  }
}

// ---------------- scatter (counting sort by species) ----------------
__global__ void scatter_kernel(const int* __restrict__ spec,
                               int* __restrict__ sorted, int* __restrict__ cursors) {
  int b = blockIdx.x * 256 + threadIdx.x;
  if (b >= NN) return;
  int pos = atomicAdd(&cursors[spec[b]], 1);
  sorted[pos] = b;
}

// ---------------- precompute per-(species,channel) GEMM matrix M ----------------
// M[e][c][row=(o*9+i)][col] ; col<45: sym pair (j,l); 45..53: linear j; 54: const; 55: pad
__global__ void build_M(const float* __restrict__ U1, const float* __restrict__ U2,
                        const float* __restrict__ U3, const float* __restrict__ W1,
                        const float* __restrict__ W2, const float* __restrict__ W3,
                        float* __restrict__ M) {
  int idx = blockIdx.x * 256 + threadIdx.x;  // exact: 10*128*96*56
  int col = idx % KCOLS; int t = idx / KCOLS;
  int row = t % KROWS;  t /= KROWS;
  int c = t % CC;       int e = t / CC;
  if (e >= SS) return;
  float v = 0.f;
  if (row < 81 && col < 55) {
    int o = row / 9, i = row % 9;
    if (col < 45) {
      int j = PJ[col], l = PL[col];
      const float* u3a = U3 + ((((o*9+i)*9 + j)*9 + l) * 23);
      const float* u3b = U3 + ((((o*9+i)*9 + l)*9 + j) * 23);
      const float* w3  = W3 + e * 23 * CC + c;
      float s = 0.f;
      for (int k = 0; k < 23; k++) {
        float u = u3a[k] + ((j != l) ? u3b[k] : 0.f);  // fold symmetric part
        s += u * w3[k * CC];
      }
      v = s;
    } else if (col < 54) {
      int j = col - 45;
      const float* u2 = U2 + (((o*9+i)*9 + j) * 8);
      const float* w2 = W2 + e * 8 * CC + c;
      float s = 0.f;
      for (int k = 0; k < 8; k++) s += u2[k] * w2[k * CC];
      v = s;
    } else {
      const float* u1 = U1 + (o*9+i) * 3;
      const float* w1 = W1 + e * 3 * CC + c;
      v = u1[0]*w1[0] + u1[1]*w1[CC] + u1[2]*w1[2*CC];
    }
  }
  M[idx] = v;
}

// ---------------- main symmetric contraction via WMMA ----------------
// grid (MAXT, C), block 32 (one wave). Per wave: 6 row-tiles x 14 k-steps = 84 WMMAs.
__global__ __launch_bounds__(32)
void sym_contract(const float* __restrict__ nf, const int* __restrict__ sorted,
                  const int* __restrict__ tspec, const int* __restrict__ tstart,
                  const int* __restrict__ tcnt, const float* __restrict__ Mfull,
                  float* __restrict__ y2) {
  int t = blockIdx.x, c = blockIdx.y;
  int cnt = tcnt[t];
  if (cnt == 0) return;
  int e = tspec[t], g0 = tstart[t];
  int lane = threadIdx.x;

  __shared__ float xs[16 * 9];      // x per node in tile
  __shared__ float F[16 * 58];      // feature matrix, F[n][col], stride 58 (conflict-free)
  __shared__ float part[2 * 16 * 9];

  for (int idx = lane; idx < 144; idx += 32) {
    int n = idx / 9, d = idx - n * 9;
    float v = 0.f;
    if (n < cnt) { int node = sorted[g0 + n]; v = nf[(node * CC + c) * DD + d]; }
    xs[idx] = v;
  }
  __syncthreads();
  for (int idx = lane; idx < 16 * KCOLS; idx += 32) {
    int n = idx & 15, col = idx >> 4;
    float v;
    if (col < 45)       v = xs[n*9 + PJ[col]] * xs[n*9 + PL[col]];
    else if (col < 54)  v = xs[n*9 + col - 45];
    else if (col == 54) v = 1.f;
    else                v = 0.f;
    F[n * 58 + col] = v;
  }
  __syncthreads();

  int half = lane >> 4, ln = lane & 15;
  const float* Mb = Mfull + (size_t)(e * CC + c) * (KROWS * KCOLS);
  v8f acc[6];
#pragma unroll
  for (int r = 0; r < 6; r++) acc[r] = (v8f){};

  for (int s = 0; s < KCOLS / 4; s++) {
    int kc = 4 * s + 2 * half;                 // even -> 8B aligned
    v2f bf;
    bf.x = F[ln * 58 + kc];
    bf.y = F[ln * 58 + kc + 1];
    const float* ap = Mb + kc;
#pragma unroll
    for (int r = 0; r < 6; r++) {
      v2f af = *(const v2f*)(ap + (r * 16 + ln) * KCOLS);  // zero-padded rows 81..95
      acc[r] = wmma4(af, bf, acc[r]);
    }
  }

  // y[o] = sum_i B[o,i] * x_i ; each lane holds rows with bit3(row)==half
#pragma unroll
  for (int o = 0; o < 9; o++) {
    float yo = 0.f;
#pragma unroll
    for (int i = 0; i < 9; i++) {
      int oi = o * 9 + i;
      int r = oi >> 4, vv = oi & 7, hn = (oi >> 3) & 1;
      float bv = (half == hn) ? acc[r][vv] : 0.f;
      yo += bv * xs[ln * 9 + i];
    }
    part[half * 144 + ln * 9 + o] = yo;
  }
  __syncthreads();
  for (int idx = lane; idx < 144; idx += 32) {
    int n = idx / 9, o = idx - n * 9;
    if (n < cnt) {
      int node = sorted[g0 + n];
      y2[(node * DD + o) * CC + c] = part[idx] + part[144 + idx];  // y stored (b,o,c)
    }
  }
}

// ---------------- e3nn Linear per l-block via WMMA + skip connection ----------------
// grid (640, 8, 3), block 32. out[b,d,m] = (1/sqrt(C)) * sum_c y[b,c,m] W_l[c,d] + sc
__global__ __launch_bounds__(32)
void linear_kernel(const float* __restrict__ y2, const float* __restrict__ lw,
                   const float* __restrict__ sc, float* __restrict__ out) {
  int l = blockIdx.z;
  int ml = (l == 0) ? 1 : (l == 1) ? 3 : 5;
  int al = (l == 0) ? 0 : (l == 1) ? 1 : 4;
  int rtiles = (NN / 16) * ml;
  int rt = blockIdx.x;
  if (rt >= rtiles) return;
  int d0 = blockIdx.y * 16;
  int lane = threadIdx.x, half = lane >> 4, ln = lane & 15;

  int row = rt * 16 + ln;
  int b = row / ml, mm = row - b * ml;
  const float* arow = y2 + (b * DD + al + mm) * CC;   // contiguous over c
  const float* wl = lw + l * CC * CC + d0 + ln;

  v8f acc = (v8f){};
  for (int s = 0; s < CC / 4; s++) {
    int c0 = 4 * s + 2 * half;
    v2f af = *(const v2f*)(arow + c0);
    v2f bf;
    bf.x = wl[c0 * CC];
    bf.y = wl[(c0 + 1) * CC];
    acc = wmma4(af, bf, acc);
  }
  const float inv = 0.08838834764831845f;  // 1/sqrt(128)
#pragma unroll
  for (int v = 0; v < 8; v++) {
    int r2 = rt * 16 + v + 8 * half;
    int b2 = r2 / ml, mm2 = r2 - b2 * ml;
    int oidx = (b2 * CC + d0 + ln) * DD + al + mm2;
    out[oidx] = acc[v] * inv + sc[oidx];
  }
}

extern "C" void kernel_launch(void* const* d_in, const int* in_sizes, int n_in,
                              void* d_out, int out_size, void* d_ws, size_t ws_size,
                              hipStream_t stream) {
  (void)in_sizes; (void)n_in; (void)out_size; (void)ws_size;
  const float* nf    = (const float*)d_in[0];  // (N,C,9)
  const float* attrs = (const float*)d_in[1];  // (N,10)
  const float* sc    = (const float*)d_in[2];  // (N,C,9)
  const float* U1    = (const float*)d_in[3];  // (9,9,3)
  const float* U2    = (const float*)d_in[4];  // (9,9,9,8)
  const float* U3    = (const float*)d_in[5];  // (9,9,9,9,23)
  const float* W1    = (const float*)d_in[6];  // (10,3,C)
  const float* W2    = (const float*)d_in[7];  // (10,8,C)
  const float* W3    = (const float*)d_in[8];  // (10,23,C)
  const float* lw    = (const float*)d_in[9];  // (3,C,C)
  float* out = (float*)d_out;

  const size_t M_ELEMS = (size_t)SS * CC * KROWS * KCOLS;   // 6,881,280 floats
  const size_t Y_ELEMS = (size_t)NN * DD * CC;              // 2,359,296 floats
  float* Mfull = (float*)d_ws;
  float* y2    = Mfull + M_ELEMS;
  int*   ibase = (int*)(y2 + Y_ELEMS);
  int* spec    = ibase;            // 2048
  int* sorted  = ibase + 2048;     // 2048
  int* counts  = ibase + 4096;
  int* cursors = ibase + 4128;
  int* tspec   = ibase + 4144;
  int* tstart  = ibase + 4304;
  int* tcnt    = ibase + 4464;

  init_kernel<<<1, 256, 0, stream>>>(ibase);
  species_kernel<<<NN / 256, 256, 0, stream>>>(attrs, spec, counts);
  scan_kernel<<<1, 32, 0, stream>>>(ibase);
  scatter_kernel<<<NN / 256, 256, 0, stream>>>(spec, sorted, cursors);
  build_M<<<(unsigned)(M_ELEMS / 256), 256, 0, stream>>>(U1, U2, U3, W1, W2, W3, Mfull);
  dim3 g1(MAXT, CC);
  sym_contract<<<g1, 32, 0, stream>>>(nf, sorted, tspec, tstart, tcnt, Mfull, y2);
  dim3 g2(640, 8, 3);
  linear_kernel<<<g2, 32, 0, stream>>>(y2, lw, sc, out);
}